// loss_generator_61967788146775
// MI455X (gfx1250) — compile-verified
//
#include <hip/hip_runtime.h>
#include <math.h>

#define B_ 2
#define T_ 9
#define C_ 4
#define H_ 512
#define W_ 512
#define DT_ 0.001f
#define DX_ (1.0f/512.0f)
#define DY_ (1.0f/512.0f)
#define GAMMA_ 1.4f
#define SF_ 4

typedef __attribute__((ext_vector_type(2))) float v2f;
typedef __attribute__((ext_vector_type(4))) float v4f;
typedef __attribute__((ext_vector_type(8))) float v8f;

__device__ __forceinline__ size_t oidx(int b, int t, int c, int h, int w) {
    return ((((size_t)(b * T_ + t)) * C_ + c) * H_ + h) * W_ + w;
}

__device__ __forceinline__ v4f ld4(const float* __restrict__ p, int b, int t, int c, int h, int w) {
    return *reinterpret_cast<const v4f*>(p + oidx(b, t, c, h, w));   // 16B aligned when w%4==0
}

// Exact 32-lane sum via V_WMMA_F32_16X16X4_F32.
// A(16x4): lane L(<16) -> row M=L, K=0; lane L(>=16) -> row M=L-16, K=2. With B=ones(4x16),
// D[M][N] = rowsum_M (independent of B lane layout). Per-lane sum of the 8 D regs gives
// sum_{M=0..7} (lanes 0-15) / sum_{M=8..15} (lanes 16-31); xor-16 swizzle-add completes the total.
// EXEC must be all ones at call site (uniform call, blockDim % 32 == 0).
__device__ __forceinline__ float wave_sum_wmma(float v) {
    v2f a; a[0] = v;    a[1] = 0.0f;
    v2f b; b[0] = 1.0f; b[1] = 1.0f;
    v8f c = {};
    v8f d = __builtin_amdgcn_wmma_f32_16x16x4_f32(
        false, a, false, b, (short)0, c, false, false);
    float s = 0.0f;
#pragma unroll
    for (int i = 0; i < 8; ++i) s += d[i];
    // SWAPX16: ds_swizzle group-of-32, xor=0x10, or=0, and=0x1f -> offset 0x401f
    s += __int_as_float(__builtin_amdgcn_ds_swizzle(__float_as_int(s), 0x401f));
    return s;
}

__device__ __forceinline__ void block_atomic_add(float v, float* dst, float* red) {
    float ws = wave_sum_wmma(v);
    int lane = threadIdx.x & 31;
    int wid  = threadIdx.x >> 5;
    if (lane == 0) red[wid] = ws;
    __syncthreads();
    if (threadIdx.x == 0) {
        float t = 0.0f;
        int nw = (blockDim.x + 31) >> 5;
        for (int i = 0; i < nw; ++i) t += red[i];
        atomicAdd(dst, t);
    }
}

__global__ void zero_kernel(float* ws) {
    if (threadIdx.x < 32) ws[threadIdx.x] = 0.0f;
}

// ---------------- TV: per-(b,t) sum over h,w in [1,512) -------------------------------
// Each thread owns 4 consecutive w (aligned float4 loads); west neighbor needs one scalar.
__global__ void tv_kernel(const float* __restrict__ out, float* __restrict__ tv) {
    __shared__ float red[8];
    int bt = blockIdx.y;                  // b*9 + t
    int b = bt / T_, t = bt % T_;
    int idx = blockIdx.x * blockDim.x + threadIdx.x;
    float acc = 0.0f;
    if (idx < 511 * 128) {
        int h  = idx / 128 + 1;           // 1..511
        int w0 = (idx % 128) * 4;         // 0,4,...,508
        // center row: U at w0-1 .. w0+3 ; north row: U at w0 .. w0+3
        float Uc[5][4], Un[4][4];
        {
            v4f R = ld4(out, b, t, 0, h, w0), U = ld4(out, b, t, 1, h, w0);
            v4f V = ld4(out, b, t, 2, h, w0), E = ld4(out, b, t, 3, h, w0);
#pragma unroll
            for (int k = 0; k < 4; ++k) {
                Uc[k + 1][0] = R[k]; Uc[k + 1][1] = U[k]; Uc[k + 1][2] = V[k];
                Uc[k + 1][3] = E[k] + 0.5f * (U[k] * U[k] + V[k] * V[k]);
            }
        }
        if (w0 > 0) {
            float r = out[oidx(b, t, 0, h, w0 - 1)], u = out[oidx(b, t, 1, h, w0 - 1)];
            float v = out[oidx(b, t, 2, h, w0 - 1)], e = out[oidx(b, t, 3, h, w0 - 1)];
            Uc[0][0] = r; Uc[0][1] = u; Uc[0][2] = v; Uc[0][3] = e + 0.5f * (u * u + v * v);
        } else {
            Uc[0][0] = Uc[0][1] = Uc[0][2] = Uc[0][3] = 0.0f;   // unused (j=0 invalid)
        }
        {
            v4f R = ld4(out, b, t, 0, h - 1, w0), U = ld4(out, b, t, 1, h - 1, w0);
            v4f V = ld4(out, b, t, 2, h - 1, w0), E = ld4(out, b, t, 3, h - 1, w0);
#pragma unroll
            for (int k = 0; k < 4; ++k) {
                Un[k][0] = R[k]; Un[k][1] = U[k]; Un[k][2] = V[k];
                Un[k][3] = E[k] + 0.5f * (U[k] * U[k] + V[k] * V[k]);
            }
        }
#pragma unroll
        for (int j = 0; j < 4; ++j) {
            int w = w0 + j;
            if (w >= 1) {
#pragma unroll
                for (int c = 0; c < 4; ++c)
                    acc += DX_ * fabsf(Uc[j + 1][c] - Un[j][c])
                         + DY_ * fabsf(Uc[j + 1][c] - Uc[j][c]);
            }
        }
    }
    block_atomic_add(acc, &tv[bt], red);
}

// ---------------- Entropy residual: LDS-tiled stencil, log once per point --------------
__global__ void ent_kernel(const float* __restrict__ out, float* __restrict__ acc) {
    __shared__ float p0[18 * 18];   // -rho*s       at time t (halo, zero-padded)
    __shared__ float p1[18 * 18];   // -rho*u*s
    __shared__ float p2[18 * 18];   // -rho*v*s
    __shared__ float red[8];
    int tid = threadIdx.x;
    int bt = blockIdx.z;            // b*8 + t
    int b = bt >> 3, t = bt & 7;
    int gy0 = blockIdx.y * 16, gx0 = blockIdx.x * 16;

    for (int i = tid; i < 18 * 18; i += 256) {
        int lh = i / 18, lw = i % 18;
        int gh = gy0 + lh - 1, gw = gx0 + lw - 1;
        float f0 = 0.0f, f1 = 0.0f, f2 = 0.0f;
        if (gh >= 0 && gh < H_ && gw >= 0 && gw < W_) {
            float rho = out[oidx(b, t, 0, gh, gw)];
            float u   = out[oidx(b, t, 1, gh, gw)];
            float v   = out[oidx(b, t, 2, gh, gw)];
            float e   = out[oidx(b, t, 3, gh, gw)];
            // p = rho*(g-1)*e  =>  s = log(p/(g*rho)) = log((g-1)/g * e)
            float s = logf((GAMMA_ - 1.0f) / GAMMA_ * e);
            f0 = -rho * s; f1 = f0 * u; f2 = f0 * v;
        }
        p0[i] = f0; p1[i] = f1; p2[i] = f2;
    }
    __syncthreads();

    int tx = tid & 15, ty = tid >> 4;
    int gh = gy0 + ty, gw = gx0 + tx;
    float rho1 = out[oidx(b, t + 1, 0, gh, gw)];
    float e1   = out[oidx(b, t + 1, 3, gh, gw)];
    float s1   = logf((GAMMA_ - 1.0f) / GAMMA_ * e1);
    float phi0n = -rho1 * s1;

    int lc = (ty + 1) * 18 + (tx + 1);
    float ent = (phi0n - p0[lc]) * (1.0f / DT_)
              + (p1[lc + 1]  - p1[lc - 1])  * (0.5f / DX_)
              + (p2[lc + 18] - p2[lc - 18]) * (0.5f / DY_);
    float r = ent > 0.0f ? ent : 0.0f;
    block_atomic_add(r * r, acc, red);
}

// ---------------- Lax-Friedrichs residual ---------------------------------------------
__device__ __forceinline__ void qf_point(float r, float u, float v, float e, float* Q, float* F) {
    float E = e + 0.5f * (u * u + v * v);
    float p = r * (GAMMA_ - 1.0f) * e;
    float rE = r * E;
    Q[0] = r;     Q[1] = r * u;         Q[2] = r * v;     Q[3] = rE;
    F[0] = r * u; F[1] = r * u * u + p; F[2] = r * u * v; F[3] = (rE + p) * u;
}
__device__ __forceinline__ void qg_point(float r, float u, float v, float e, float* Q, float* G) {
    float E = e + 0.5f * (u * u + v * v);
    float p = r * (GAMMA_ - 1.0f) * e;
    float rE = r * E;
    Q[0] = r;     Q[1] = r * u;     Q[2] = r * v;         Q[3] = rE;
    G[0] = r * v; G[1] = r * u * v; G[2] = r * v * v + p; G[3] = (rE + p) * v;
}

// Each thread owns one b, one h in [1,510], a 4-wide aligned w-group.
__global__ void fq_kernel(const float* __restrict__ out, float* __restrict__ acc) {
    __shared__ float red[8];
    int b = blockIdx.y;
    int idx = blockIdx.x * blockDim.x + threadIdx.x;
    float a = 0.0f;
    if (idx < 510 * 128) {
        int h  = idx / 128 + 1;           // 1..510
        int w0 = (idx % 128) * 4;         // 0,4,...,508
        // primitives at t=8: center row w0-1..w0+4 (idx k = w-(w0-1)), rows h+-1 at w0..w0+3
        float r[6], u[6], v[6], e[6];
        float rn[4], un[4], vn[4], en[4], rs[4], us[4], vs[4], es[4];
        {
            v4f R = ld4(out, b, 8, 0, h, w0), U = ld4(out, b, 8, 1, h, w0);
            v4f V = ld4(out, b, 8, 2, h, w0), E = ld4(out, b, 8, 3, h, w0);
#pragma unroll
            for (int k = 0; k < 4; ++k) { r[k+1]=R[k]; u[k+1]=U[k]; v[k+1]=V[k]; e[k+1]=E[k]; }
        }
        if (w0 > 0) {
            r[0] = out[oidx(b, 8, 0, h, w0 - 1)]; u[0] = out[oidx(b, 8, 1, h, w0 - 1)];
            v[0] = out[oidx(b, 8, 2, h, w0 - 1)]; e[0] = out[oidx(b, 8, 3, h, w0 - 1)];
        } else { r[0] = u[0] = v[0] = e[0] = 0.0f; }
        if (w0 + 4 < W_) {
            r[5] = out[oidx(b, 8, 0, h, w0 + 4)]; u[5] = out[oidx(b, 8, 1, h, w0 + 4)];
            v[5] = out[oidx(b, 8, 2, h, w0 + 4)]; e[5] = out[oidx(b, 8, 3, h, w0 + 4)];
        } else { r[5] = u[5] = v[5] = e[5] = 0.0f; }
        {
            v4f R = ld4(out, b, 8, 0, h - 1, w0), U = ld4(out, b, 8, 1, h - 1, w0);
            v4f V = ld4(out, b, 8, 2, h - 1, w0), E = ld4(out, b, 8, 3, h - 1, w0);
#pragma unroll
            for (int k = 0; k < 4; ++k) { rn[k]=R[k]; un[k]=U[k]; vn[k]=V[k]; en[k]=E[k]; }
        }
        {
            v4f R = ld4(out, b, 8, 0, h + 1, w0), U = ld4(out, b, 8, 1, h + 1, w0);
            v4f V = ld4(out, b, 8, 2, h + 1, w0), E = ld4(out, b, 8, 3, h + 1, w0);
#pragma unroll
            for (int k = 0; k < 4; ++k) { rs[k]=R[k]; us[k]=U[k]; vs[k]=V[k]; es[k]=E[k]; }
        }

        float S[4][4];
        bool valid[4];
#pragma unroll
        for (int j = 0; j < 4; ++j) {
            int w = w0 + j;
            valid[j] = (w >= 1 && w <= 510);
            if (valid[j]) {
                float Qm[4], Fm[4], Qc[4], Fc[4], Qp[4], Fp[4];
                qf_point(r[j],     u[j],     v[j],     e[j],     Qm, Fm);   // w-1
                qf_point(r[j + 1], u[j + 1], v[j + 1], e[j + 1], Qc, Fc);   // w
                qf_point(r[j + 2], u[j + 2], v[j + 2], e[j + 2], Qp, Fp);   // w+1
                float Qn[4], Gn[4], Qs[4], Gs[4];
                qg_point(rn[j], un[j], vn[j], en[j], Qn, Gn);               // h-1
                qg_point(rs[j], us[j], vs[j], es[j], Qs, Gs);               // h+1
#pragma unroll
                for (int c = 0; c < 4; ++c) {
                    S[j][c] = (Fp[c] - Fm[c]) * (0.5f / DX_)
                            - (0.5f / DT_) * (Qp[c] - 2.0f * Qc[c] + Qm[c])
                            + (Gs[c] - Gn[c]) * (0.5f / DY_)
                            - (0.5f / DT_) * (Qs[c] - 2.0f * Qc[c] + Qn[c]);
                }
            } else {
#pragma unroll
                for (int c = 0; c < 4; ++c) S[j][c] = 0.0f;
            }
        }

        float qp[4][4];
        {
            v4f R = ld4(out, b, 0, 0, h, w0), U = ld4(out, b, 0, 1, h, w0);
            v4f V = ld4(out, b, 0, 2, h, w0), E = ld4(out, b, 0, 3, h, w0);
#pragma unroll
            for (int j = 0; j < 4; ++j) {
                qp[j][0] = R[j]; qp[j][1] = R[j] * U[j]; qp[j][2] = R[j] * V[j];
                qp[j][3] = R[j] * (E[j] + 0.5f * (U[j] * U[j] + V[j] * V[j]));
            }
        }
        for (int t = 1; t < T_; ++t) {
            if (t < T_ - 1) __builtin_prefetch(&out[oidx(b, t + 1, 0, h, w0)], 0, 1);
            v4f R = ld4(out, b, t, 0, h, w0), U = ld4(out, b, t, 1, h, w0);
            v4f V = ld4(out, b, t, 2, h, w0), E = ld4(out, b, t, 3, h, w0);
#pragma unroll
            for (int j = 0; j < 4; ++j) {
                float q0 = R[j], q1 = R[j] * U[j], q2 = R[j] * V[j];
                float q3 = R[j] * (E[j] + 0.5f * (U[j] * U[j] + V[j] * V[j]));
                if (valid[j]) {
                    float d0 = (q0 - qp[j][0]) * (1.0f / DT_) + S[j][0];
                    float d1 = (q1 - qp[j][1]) * (1.0f / DT_) + S[j][1];
                    float d2 = (q2 - qp[j][2]) * (1.0f / DT_) + S[j][2];
                    float d3 = (q3 - qp[j][3]) * (1.0f / DT_) + S[j][3];
                    a += d0 * d0 + d1 * d1 + d2 * d2 + d3 * d3;
                }
                qp[j][0] = q0; qp[j][1] = q1; qp[j][2] = q2; qp[j][3] = q3;
            }
        }
    }
    block_atomic_add(a, acc, red);
}

// ---------------- 4x4 average pool + squared diff vs low-res input ---------------------
__global__ void pool_kernel(const float* __restrict__ out, const float* __restrict__ inlr,
                            float* __restrict__ acc) {
    __shared__ float red[8];
    const int PH = H_ / SF_, PW = W_ / SF_;
    int idx = blockIdx.x * blockDim.x + threadIdx.x;
    float a = 0.0f;
    if (idx < B_ * T_ * C_ * PH * PW) {
        int pw = idx % PW; int tmp = idx / PW;
        int ph = tmp % PH; tmp /= PH;
        int c  = tmp % C_; tmp /= C_;
        int t  = tmp % T_;
        int b  = tmp / T_;
        float s = 0.0f;
#pragma unroll
        for (int i = 0; i < SF_; ++i) {
            v4f row = ld4(out, b, t, c, ph * SF_ + i, pw * SF_);
            s += row[0] + row[1] + row[2] + row[3];
        }
        s *= (1.0f / 16.0f);
        float d = inlr[idx] - s;
        a = d * d;
    }
    block_atomic_add(a, acc, red);
}

// ---------------- Finalize -------------------------------------------------------------
__global__ void finalize_kernel(const float* __restrict__ ws, const float* __restrict__ regw,
                                float* __restrict__ outp) {
    if (threadIdx.x == 0) {
        const float* tv = ws + 3;
        float tvl = 0.0f;
        for (int b = 0; b < B_; ++b)
            for (int t = 1; t < T_; ++t) {
                float d = tv[b * T_ + t] - tv[b * T_ + t - 1];
                d = d > 0.0f ? d : 0.0f;
                tvl += d * d;
            }
        tvl *= (1.0f / 16.0f);
        float entl = ws[0] / (float)(2.0 * 8.0 * 512.0 * 512.0);
        float fql  = ws[1] / (float)(2.0 * 8.0 * 4.0 * 510.0 * 510.0);
        float reg  = regw[0] * sqrtf(ws[2]);
        outp[0] = fql + reg + 0.01f * tvl + 0.01f * entl;
        outp[1] = reg;
    }
}

extern "C" void kernel_launch(void* const* d_in, const int* in_sizes, int n_in,
                              void* d_out, int out_size, void* d_ws, size_t ws_size,
                              hipStream_t stream) {
    const float* inlr = (const float*)d_in[0];   // [2,9,4,128,128]
    const float* out  = (const float*)d_in[1];   // [2,9,4,512,512]
    const float* regw = (const float*)d_in[4];   // scalar
    float* ws  = (float*)d_ws;                   // [0]=ent [1]=fq [2]=reg [3..20]=TV
    float* res = (float*)d_out;                  // 2 floats: loss, reg_loss

    zero_kernel<<<1, 64, 0, stream>>>(ws);
    tv_kernel<<<dim3((511 * 128 + 255) / 256, B_ * T_), 256, 0, stream>>>(out, ws + 3);
    ent_kernel<<<dim3(32, 32, B_ * 8), 256, 0, stream>>>(out, ws + 0);
    fq_kernel<<<dim3((510 * 128 + 255) / 256, B_), 256, 0, stream>>>(out, ws + 1);
    pool_kernel<<<dim3((B_ * T_ * C_ * (H_ / SF_) * (W_ / SF_) + 255) / 256), 256, 0, stream>>>(out, inlr, ws + 2);
    finalize_kernel<<<1, 32, 0, stream>>>(ws, regw, res);
}